// ChunkedSourceCompressor_84430467105068
// MI455X (gfx1250) — compile-verified
//
#include <hip/hip_runtime.h>

// ---------------------------------------------------------------------------
// ChunkedSourceCompressor for MI455X (gfx1250), fp32 path with V_WMMA_F32_16X16X4_F32.
// Memory-bound problem (~545 MB HBM traffic -> ~23 us at 23.3 TB/s); keys/values
// are never materialized to HBM: fused per-chunk in LDS.
// Workspace layout (floats): gavg[B*D] | q[B*K*D] | r[B*D] | cs[B*K*NB] | blk[B*NB*K*D]
//   total ~139 MB (assumed <= ws_size).
// Wave-uniform tile scheduling is routed through readfirstlane so all control
// flow around the WMMAs is scalar (EXEC stays all-ones, per ISA 7.12 rules).
// ---------------------------------------------------------------------------

#define BATCH 32
#define NTOK  16384
#define DIM   65
#define CHUNK 64
#define KQ    64
#define NB    (NTOK / CHUNK)   // 256
#define DPAD  68               // K-dim pad (multiple of 4), rows/cols 65..67 zeroed
#define NPAD  80               // N-dim pad (multiple of 16), excess discarded by lane guard
#define SCALE 0.124034734589208f   // 65^-0.5

typedef __attribute__((ext_vector_type(2))) float v2f;
typedef __attribute__((ext_vector_type(8))) float v8f;

__device__ __forceinline__ float gelu_f(float v) {
  // exact GELU: 0.5*x*(1+erf(x/sqrt(2)))
  return 0.5f * v * (1.0f + erff(v * 0.70710678118654752440f));
}

// One 16x16 f32 output tile of C = A(MxK) * B(KxN), K = 4*ksteps, via
// V_WMMA_F32_16X16X4_F32.  Fragment layouts per CDNA5 ISA 7.12.2:
//   A 16x4 : lane m = lane&15, K = vgpr + 2*(lane>>4)
//   B 4x16 : lane n = lane&15, K = vgpr + 2*(lane>>4)
//   C 16x16: vgpr r -> M = r + 8*(lane>>4), N = lane&15
template <int LDA, int LDB>
__device__ __forceinline__ v8f wmma_f32_tile(const float (*A)[LDA], const float (*B)[LDB],
                                             int m0, int n0, int ksteps, int lm, int kg) {
  v8f acc = {};
  const int ka = 2 * kg;
  for (int ks = 0; ks < ksteps; ++ks) {
    const int k0 = 4 * ks;
    v2f a, b;
    a[0] = A[m0 + lm][k0 + ka + 0];
    a[1] = A[m0 + lm][k0 + ka + 1];
    b[0] = B[k0 + ka + 0][n0 + lm];
    b[1] = B[k0 + ka + 1][n0 + lm];
    acc = __builtin_amdgcn_wmma_f32_16x16x4_f32(false, a, false, b, (short)0, acc,
                                                false, false);
  }
  return acc;
}

// --------------------------- K1: global average ----------------------------
// grid (NTOK/256, BATCH) x 256 threads; partial sums via LDS then global atomics.
__global__ __launch_bounds__(256) void k1_gavg(const float* __restrict__ x,
                                               float* __restrict__ gavg) {
  __shared__ float acc[DIM];
  const int tid = threadIdx.x;
  const int b = blockIdx.y;
  if (tid < DIM) acc[tid] = 0.0f;
  __syncthreads();
  const float* xp = x + ((size_t)b * NTOK + (size_t)blockIdx.x * 256) * DIM;
  for (int i = tid; i < 256 * DIM; i += 256) atomicAdd(&acc[i % DIM], xp[i]);  // ds_add_f32
  __syncthreads();
  if (tid < DIM) atomicAdd(&gavg[b * DIM + tid], acc[tid]);                    // global_atomic_add_f32
}

// --------------------------- K2: query MLP + residual proj -----------------
// grid BATCH x 256.  q[b,k,d] = (gelu(g@Wq1+bq1)@Wq2+bq2)*SCALE ; r[b,d]=g@Wr+br
__global__ __launch_bounds__(256) void k2_qgen(const float* __restrict__ gavg_sum,
                                               const float* __restrict__ Wq1,
                                               const float* __restrict__ bq1,
                                               const float* __restrict__ Wq2,
                                               const float* __restrict__ bq2,
                                               const float* __restrict__ Wr,
                                               const float* __restrict__ br,
                                               float* __restrict__ qout,
                                               float* __restrict__ rout) {
  __shared__ float g[DIM], h[DIM];
  const int b = blockIdx.x, tid = threadIdx.x;
  if (tid < DIM) g[tid] = gavg_sum[b * DIM + tid] * (1.0f / (float)NTOK);
  __syncthreads();
  if (tid < DIM) {
    float a = bq1[tid], rr = br[tid];
    for (int i = 0; i < DIM; ++i) {
      a += g[i] * Wq1[i * DIM + tid];
      rr += g[i] * Wr[i * DIM + tid];
    }
    h[tid] = gelu_f(a);
    rout[b * DIM + tid] = rr;
  }
  __syncthreads();
  for (int j = tid; j < KQ * DIM; j += 256) {
    float a = bq2[j];
    for (int i = 0; i < DIM; ++i) a += h[i] * Wq2[i * (KQ * DIM) + j];
    qout[(size_t)b * KQ * DIM + j] = a * SCALE;
  }
}

// --------------------------- K3: fused per-chunk attention -----------------
// grid (NB/4, BATCH) x 256 (8 waves).  Each WG: 4 chunks of one batch.
__global__ __launch_bounds__(256) void k3_attn(const float* __restrict__ x,
                                               const float* __restrict__ Wk,
                                               const float* __restrict__ bk,
                                               const float* __restrict__ Wv,
                                               const float* __restrict__ bv,
                                               const float* __restrict__ pos,
                                               const float* __restrict__ cross_q,
                                               const float* __restrict__ q,
                                               float* __restrict__ blk_out,  // [B,NB,K,D]
                                               float* __restrict__ cs_out) { // [B,K,NB]
  __shared__ float xs[CHUNK][DPAD];        // x tile, cols 65..67 = 0
  __shared__ float Wks[DPAD][NPAD];        // Wk padded (zero-filled first)
  __shared__ float Wvs[DPAD][NPAD];
  __shared__ float qs[KQ][DPAD];           // q rows (already *SCALE), cols 65..67 = 0
  __shared__ float keysT[DPAD][CHUNK];     // keys transposed [d][c]; rows 65..67 = 0
  __shared__ float vals[CHUNK][NPAD];      // values [c][d]
  __shared__ float sc[KQ][CHUNK + 1];      // scores / softmax weights (pad: no bank conflicts)
  __shared__ float bks[DIM], bvs[DIM], cqs[DIM];
  __shared__ float cs_s[KQ];

  const int b = blockIdx.y;
  const int cg = blockIdx.x;
  const int tid = threadIdx.x;
  // wave index is uniform within a wave; route through readfirstlane so the
  // compiler emits scalar branches (EXEC untouched around all WMMAs).
  const int wave = __builtin_amdgcn_readfirstlane(tid >> 5);
  const int lane = tid & 31;
  const int lm = lane & 15, kg = lane >> 4;

  // ---- one-time per-WG loads --------------------------------------------
  for (int i = tid; i < DPAD * NPAD; i += 256) { (&Wks[0][0])[i] = 0.0f; (&Wvs[0][0])[i] = 0.0f; }
  for (int i = tid; i < KQ * DPAD; i += 256) (&qs[0][0])[i] = 0.0f;
  __syncthreads();
  for (int i = tid; i < DIM * DIM; i += 256) {
    const int r = i / DIM, c = i % DIM;
    Wks[r][c] = Wk[i];
    Wvs[r][c] = Wv[i];
  }
  for (int i = tid; i < KQ * DIM; i += 256) qs[i / DIM][i % DIM] = q[(size_t)b * KQ * DIM + i];
  if (tid < DIM) { bks[tid] = bk[tid]; bvs[tid] = bv[tid]; cqs[tid] = cross_q[tid]; }
  if (tid < (DPAD - DIM) * CHUNK) keysT[DIM + tid / CHUNK][tid % CHUNK] = 0.0f;
  __syncthreads();

  for (int ci = 0; ci < 4; ++ci) {
    const int n = cg * 4 + ci;
    const float* xt = x + ((size_t)b * NTOK + (size_t)n * CHUNK) * DIM;

    // ---- stage x tile into LDS; prefetch next tile (global_prefetch_b8) --
    for (int i = tid; i < CHUNK * DIM; i += 256) xs[i / DIM][i % DIM] = xt[i];
    for (int i = tid; i < CHUNK * (DPAD - DIM); i += 256)
      xs[i / (DPAD - DIM)][DIM + i % (DPAD - DIM)] = 0.0f;
    if (tid < KQ) cs_s[tid] = 0.0f;
    if (ci < 3 && tid < (CHUNK * DIM * 4 + 127) / 128)
      __builtin_prefetch((const char*)(xt + CHUNK * DIM) + tid * 128, 0, 0);
    __syncthreads();

    // ---- keys & values: gelu(x@W + b) + pos; 40 16x16 tiles over 8 waves -
    for (int t = wave; t < 40; t += 8) {
      const int tt = t % 20;
      const int mt = tt / 5, nt = tt % 5;
      v8f acc;
      if (t < 20) acc = wmma_f32_tile<DPAD, NPAD>(xs, Wks, mt * 16, nt * 16, DPAD / 4, lm, kg);
      else        acc = wmma_f32_tile<DPAD, NPAD>(xs, Wvs, mt * 16, nt * 16, DPAD / 4, lm, kg);
      for (int r = 0; r < 8; ++r) {
        const int m = mt * 16 + r + 8 * kg;  // token in chunk
        const int nn = nt * 16 + lm;         // feature d
        if (nn < DIM) {
          const float v = gelu_f(acc[r] + (t < 20 ? bks[nn] : bvs[nn])) + pos[m * DIM + nn];
          if (t < 20) keysT[nn][m] = v;
          else        vals[m][nn] = v;
        }
      }
    }
    __syncthreads();

    // ---- scores = q @ keys^T (K-dim = D, padded 68): 16 tiles ------------
    for (int t = wave; t < 16; t += 8) {
      const int mt = t / 4, nt = t % 4;
      v8f acc = wmma_f32_tile<DPAD, CHUNK>(qs, keysT, mt * 16, nt * 16, DPAD / 4, lm, kg);
      for (int r = 0; r < 8; ++r)
        sc[mt * 16 + r + 8 * kg][nt * 16 + lm] = acc[r] * SCALE;
    }
    __syncthreads();

    // ---- row softmax over chunk dim (64 rows, 1 thread/row) --------------
    if (tid < KQ) {
      float mx = -3.402823466e38f;
      for (int c = 0; c < CHUNK; ++c) mx = fmaxf(mx, sc[tid][c]);
      float s = 0.0f;
      for (int c = 0; c < CHUNK; ++c) { const float e = expf(sc[tid][c] - mx); sc[tid][c] = e; s += e; }
      const float inv = 1.0f / s;
      for (int c = 0; c < CHUNK; ++c) sc[tid][c] *= inv;
    }
    __syncthreads();

    // ---- blk = softmax @ values (K-dim = 64): 20 tiles; emit blk + cs ----
    float* bo = blk_out + ((size_t)b * NB + n) * KQ * DIM;
    for (int t = wave; t < 20; t += 8) {
      const int mt = t / 5, nt = t % 5;
      v8f acc = wmma_f32_tile<CHUNK + 1, NPAD>(sc, vals, mt * 16, nt * 16, CHUNK / 4, lm, kg);
      for (int r = 0; r < 8; ++r) {
        const int k = mt * 16 + r + 8 * kg;
        const int dd = nt * 16 + lm;
        if (dd < DIM) {
          bo[k * DIM + dd] = acc[r];
          atomicAdd(&cs_s[k], acc[r] * cqs[dd]);  // ds_add_f32
        }
      }
    }
    __syncthreads();
    if (tid < KQ) cs_out[((size_t)b * KQ + tid) * NB + n] = cs_s[tid] * SCALE;
    __syncthreads();
  }
}

// --------------------------- K4: cross-block softmax + LN ------------------
// grid (KQ, BATCH) x 128
__global__ __launch_bounds__(128) void k4_cross(const float* __restrict__ blk,
                                                const float* __restrict__ cs,
                                                const float* __restrict__ rbuf,
                                                const float* __restrict__ gamma,
                                                const float* __restrict__ beta,
                                                float* __restrict__ out) {
  __shared__ float w[NB];
  __shared__ float red[128];
  __shared__ float comp[DIM];
  __shared__ float stats[2];
  const int b = blockIdx.y, k = blockIdx.x, tid = threadIdx.x;
  const float* csr = cs + ((size_t)b * KQ + k) * NB;

  float mx = -3.402823466e38f;
  for (int i = tid; i < NB; i += 128) mx = fmaxf(mx, csr[i]);
  red[tid] = mx;
  __syncthreads();
  for (int s = 64; s > 0; s >>= 1) { if (tid < s) red[tid] = fmaxf(red[tid], red[tid + s]); __syncthreads(); }
  const float m = red[0];
  __syncthreads();
  float sm = 0.0f;
  for (int i = tid; i < NB; i += 128) { const float e = expf(csr[i] - m); w[i] = e; sm += e; }
  red[tid] = sm;
  __syncthreads();
  for (int s = 64; s > 0; s >>= 1) { if (tid < s) red[tid] += red[tid + s]; __syncthreads(); }
  const float inv = 1.0f / red[0];
  __syncthreads();

  if (tid < DIM) {
    const float* bp = blk + ((size_t)b * NB * KQ + k) * DIM + tid;
    float a = 0.0f;
    for (int n = 0; n < NB; ++n) a += w[n] * bp[(size_t)n * KQ * DIM];
    comp[tid] = a * inv + rbuf[b * DIM + tid];
  }
  __syncthreads();
  if (tid == 0) {
    float s1 = 0.0f, s2 = 0.0f;
    for (int d = 0; d < DIM; ++d) { s1 += comp[d]; s2 += comp[d] * comp[d]; }
    const float mu = s1 / (float)DIM;
    stats[0] = mu;
    stats[1] = rsqrtf(s2 / (float)DIM - mu * mu + 1e-5f);
  }
  __syncthreads();
  if (tid < DIM)
    out[((size_t)b * KQ + k) * DIM + tid] = (comp[tid] - stats[0]) * stats[1] * gamma[tid] + beta[tid];
}

// ---------------------------------------------------------------------------
extern "C" void kernel_launch(void* const* d_in, const int* in_sizes, int n_in,
                              void* d_out, int out_size, void* d_ws, size_t ws_size,
                              hipStream_t stream) {
  (void)in_sizes; (void)n_in; (void)out_size; (void)ws_size;
  const float* x     = (const float*)d_in[0];
  const float* Wq1   = (const float*)d_in[1];
  const float* bq1   = (const float*)d_in[2];
  const float* Wq2   = (const float*)d_in[3];
  const float* bq2   = (const float*)d_in[4];
  const float* Wk    = (const float*)d_in[5];
  const float* bk    = (const float*)d_in[6];
  const float* Wv    = (const float*)d_in[7];
  const float* bv    = (const float*)d_in[8];
  const float* cq    = (const float*)d_in[9];
  const float* pos   = (const float*)d_in[10];
  const float* Wr    = (const float*)d_in[11];
  const float* br    = (const float*)d_in[12];
  const float* gamma = (const float*)d_in[13];
  const float* beta  = (const float*)d_in[14];

  float* ws = (float*)d_ws;
  size_t o = 0;
  float* gavg = ws + o; o += BATCH * DIM;                    // 2080
  float* qbuf = ws + o; o += (size_t)BATCH * KQ * DIM;       // 133120
  float* rbuf = ws + o; o += BATCH * DIM;                    // 2080
  float* csb  = ws + o; o += (size_t)BATCH * KQ * NB;        // 524288
  o = (o + 255) & ~(size_t)255;
  float* blk  = ws + o;                                      // B*NB*KQ*DIM = 34078720

  hipMemsetAsync(gavg, 0, BATCH * DIM * sizeof(float), stream);
  k1_gavg<<<dim3(NTOK / 256, BATCH), 256, 0, stream>>>(x, gavg);
  k2_qgen<<<BATCH, 256, 0, stream>>>(gavg, Wq1, bq1, Wq2, bq2, Wr, br, qbuf, rbuf);
  k3_attn<<<dim3(NB / 4, BATCH), 256, 0, stream>>>(x, Wk, bk, Wv, bv, pos, cq, qbuf, blk, csb);
  k4_cross<<<dim3(KQ, BATCH), 128, 0, stream>>>(blk, csb, rbuf, gamma, beta, (float*)d_out);
}